// Translation_14199161880926
// MI455X (gfx1250) — compile-verified
//
#include <hip/hip_runtime.h>
#include <stdint.h>

#define AS1 __attribute__((address_space(1)))
#define AS3 __attribute__((address_space(3)))

typedef int v4i __attribute__((vector_size(16)));

#define C_DIM 128
#define WAVES_PER_BLOCK 8
#define BLOCK_THREADS (WAVES_PER_BLOCK * 32)
#define MAX_BLOCKS 4096

// One wave moves one 512B pixel row: each of the 32 lanes async-copies 16B
// straight from global memory into LDS (no VGPR staging). Tracked by ASYNCcnt.
__device__ __forceinline__ void async_load_row16(const float* g, AS3 float* l) {
#if __has_builtin(__builtin_amdgcn_global_load_async_to_lds_b128)
  __builtin_amdgcn_global_load_async_to_lds_b128((v4i*)g, (AS3 v4i*)l, 0, 0);
#else
  uint32_t loff = (uint32_t)(uintptr_t)l;  // AS3 ptr -> LDS byte offset
  asm volatile("global_load_async_to_lds_b128 %0, %1, off"
               :
               : "v"(loff), "v"(g)
               : "memory");
#endif
}

template <int N>
__device__ __forceinline__ void wait_asynccnt() {
#if __has_builtin(__builtin_amdgcn_s_wait_asynccnt)
  __builtin_amdgcn_s_wait_asynccnt(N);
#else
  asm volatile("s_wait_asynccnt %0" : : "i"(N) : "memory");
#endif
}

__global__ __launch_bounds__(BLOCK_THREADS)
void channel_shift_async_kernel(const float* __restrict__ x,
                                const float* __restrict__ frame_value,
                                float* __restrict__ out,
                                int npix) {
  // Per-wave double buffer: 2 x 128 floats (512B) per wave, 8KB per block.
  __shared__ float tile[WAVES_PER_BLOCK][2][C_DIM];

  const int tid    = threadIdx.x;
  const int wave   = tid >> 5;
  const int lane   = tid & 31;
  const int gwave  = blockIdx.x * WAVES_PER_BLOCK + wave;
  const int nwaves = gridDim.x * WAVES_PER_BLOCK;

  // mag = trunc(15 * frame_value), matches astype(int32) truncation.
  const int mag = (int)(15.0f * frame_value[0]);

  AS3 float* lbuf0 = (AS3 float*)&tile[wave][0][lane * 4];
  AS3 float* lbuf1 = (AS3 float*)&tile[wave][1][lane * 4];

  int p = gwave;
  if (p < npix)
    async_load_row16(x + (size_t)p * C_DIM + (size_t)(lane * 4), lbuf0);

  int buf = 0;
  for (; p < npix; p += nwaves) {
    const int pn = p + nwaves;
    if (pn < npix) {
      // Prefetch next row into the other buffer, then wait for the older one.
      async_load_row16(x + (size_t)pn * C_DIM + (size_t)(lane * 4),
                       buf ? lbuf0 : lbuf1);
      wait_asynccnt<1>();
    } else {
      wait_asynccnt<0>();
    }

    // Shifted, edge-clamped gather out of LDS. Only the lower clamp can fire
    // (mag in [0,14]), so max(idx,0) is exact.
    const int c0 = lane * 4 - mag;
    float4 v;
    v.x = tile[wave][buf][c0     < 0 ? 0 : c0];
    v.y = tile[wave][buf][c0 + 1 < 0 ? 0 : c0 + 1];
    v.z = tile[wave][buf][c0 + 2 < 0 ? 0 : c0 + 2];
    v.w = tile[wave][buf][c0 + 3 < 0 ? 0 : c0 + 3];

    // 16B-aligned vector store: p*512B + lane*16B.
    *(float4*)(out + (size_t)p * C_DIM + (size_t)(lane * 4)) = v;

    buf ^= 1;
  }
}

extern "C" void kernel_launch(void* const* d_in, const int* in_sizes, int n_in,
                              void* d_out, int out_size, void* d_ws, size_t ws_size,
                              hipStream_t stream) {
  (void)in_sizes; (void)n_in; (void)d_ws; (void)ws_size;

  const float* x  = (const float*)d_in[0];   // [1,1024,1024,128] fp32
  const float* fv = (const float*)d_in[1];   // scalar frame_value
  float* out      = (float*)d_out;

  const int npix = out_size / C_DIM;         // 1024*1024 pixel rows
  int blocks = (npix + WAVES_PER_BLOCK - 1) / WAVES_PER_BLOCK;
  if (blocks > MAX_BLOCKS) blocks = MAX_BLOCKS;
  if (blocks < 1) blocks = 1;

  channel_shift_async_kernel<<<dim3(blocks), dim3(BLOCK_THREADS), 0, stream>>>(
      x, fv, out, npix);
}